// DifferentiableTransformer_53815940219302
// MI455X (gfx1250) — compile-verified
//
#include <hip/hip_runtime.h>

// ---------------- problem constants (match reference) ----------------
#define GRID_DIM   128
#define GRID_SHIFT 7                  // log2(128)
#define GRID_VOX   (GRID_DIM*GRID_DIM*GRID_DIM)   // 2^21
#define LBOX       13                 // candidate box extent per dim
#define LBOX2      (LBOX*LBOX)        // 169
#define PBOX       (LBOX*LBOX*LBOX)   // 2197
#define RMAX2      9.0f               // RMAX^2, RMAX=3.0
#define RSTEP      0.01f
#define BLOCK      256
#define MAX_RADIAL 512                // >= 302, LDS staging buffer

// CDNA5 async global->LDS DMA (ASYNCcnt path); guarded so host pass and
// older toolchains still compile (fallback = plain load + ds_store).
#if defined(__HIP_DEVICE_COMPILE__)
#if __has_builtin(__builtin_amdgcn_global_load_async_to_lds_b32) && \
    __has_builtin(__builtin_amdgcn_s_wait_asynccnt)
#define USE_ASYNC_LDS 1
#endif
#endif

// Builtin signature (from hipcc diagnostic): global param is
// addrspace(1) int* (printed "__device__ int *"), LDS side addrspace(3).
typedef __attribute__((address_space(1))) int gint_t;  // global int
typedef __attribute__((address_space(3))) int sint_t;  // LDS int

__global__ __launch_bounds__(BLOCK)
void splat_density_kernel(const float* __restrict__ coords,        // (B,N,3)
                          const unsigned char* __restrict__ active, // (B,N) bool8
                          const float* __restrict__ occ,           // (B,N)
                          const float* __restrict__ lmax,          // (3,)
                          const float* __restrict__ radial,        // (B,N,n_radial)
                          const float* __restrict__ gmat,          // (3,3)
                          float* __restrict__ out,                 // (B, 128^3)
                          int N, int n_radial)
{
    const int atom = blockIdx.x;               // flat (b*N + n)
    if (!active[atom]) return;                 // uniform per block

    __shared__ float s_rad[MAX_RADIAL];

    // ---- stage this atom's radial table into LDS ----
    const float* grow = radial + (size_t)atom * (size_t)n_radial;
#ifdef USE_ASYNC_LDS
    for (int i = threadIdx.x; i < n_radial; i += BLOCK) {
        __builtin_amdgcn_global_load_async_to_lds_b32(
            (gint_t*)(grow + i), (sint_t*)(&s_rad[i]), 0, 0);
    }
    __builtin_amdgcn_s_wait_asynccnt(0);   // my wave's DMAs into LDS landed
    __syncthreads();                        // everyone's landed
#else
    for (int i = threadIdx.x; i < n_radial; i += BLOCK) {
        s_rad[i] = grow[i];
    }
    __syncthreads();
#endif

    // ---- per-atom uniform scalars (blockIdx-uniform -> SMEM loads) ----
    const float ca = coords[3 * atom + 0];
    const float cb = coords[3 * atom + 1];
    const float cc = coords[3 * atom + 2];
    const float la = lmax[0], lb = lmax[1], lc = lmax[2];

    const float aminf = ceilf(ca - la), amaxf = floorf(ca + la);
    const float bminf = ceilf(cb - lb), bmaxf = floorf(cb + lb);
    const float cminf = ceilf(cc - lc), cmaxf = floorf(cc + lc);

    const float g00 = gmat[0], g01 = gmat[1], g02 = gmat[2];
    const float g11 = gmat[4], g12 = gmat[5];
    const float g22 = gmat[8];
    const float o   = occ[atom];

    const int   b        = atom / N;
    float* __restrict__ base_out = out + ((size_t)b << 21);  // b * 128^3
    const int   max_idx  = n_radial - 1;

    // ---- sweep the candidate box; oa fastest so adjacent lanes hit
    //      adjacent grid addresses (atomic packing in L2 lines) ----
    for (int p = threadIdx.x; p < PBOX; p += BLOCK) {
        const int oc  = p / LBOX2;
        const int rem = p - oc * LBOX2;
        const int ob  = rem / LBOX;
        const int oa  = rem - ob * LBOX;

        const float gc = cminf + (float)oc;
        const float gb = bminf + (float)ob;
        const float ga = aminf + (float)oa;

        const bool inbox = (gc <= cmaxf) & (gb <= bmaxf) & (ga <= amaxf);

        const float dc = cc - gc;
        const float db = cb - gb;
        const float da = ca - ga;
        const float dz = g22 * dc;
        const float dy = g12 * dc + g11 * db;
        const float dx = g02 * dc + g01 * db + g00 * da;
        const float d2 = dx * dx + dy * dy + dz * dz;

        if (inbox && (d2 <= RMAX2)) {
            const float r     = sqrtf(d2);
            const float rad_c = r / RSTEP;
            int ilo = (int)floorf(rad_c);
            ilo = ilo < 0 ? 0 : (ilo > max_idx ? max_idx : ilo);
            int ihi = ilo + 1;
            ihi = ihi > max_idx ? max_idx : ihi;
            const float w_hi = rad_c - (float)ilo;
            const float w_lo = 1.0f - w_hi;

            const float val = o * (w_lo * s_rad[ilo] + w_hi * s_rad[ihi]);

            // periodic wrap: & 127 == jnp.remainder(x, 128) for our range
            const int ci = ((int)gc) & (GRID_DIM - 1);
            const int bi = ((int)gb) & (GRID_DIM - 1);
            const int ai = ((int)ga) & (GRID_DIM - 1);
            const int flat = (ci << (2 * GRID_SHIFT)) + (bi << GRID_SHIFT) + ai;

            // no-return relaxed agent-scope fp32 atomic -> global_atomic_add_f32
            (void)__hip_atomic_fetch_add(&base_out[flat], val,
                                         __ATOMIC_RELAXED,
                                         __HIP_MEMORY_SCOPE_AGENT);
        }
    }
}

extern "C" void kernel_launch(void* const* d_in, const int* in_sizes, int n_in,
                              void* d_out, int out_size, void* d_ws, size_t ws_size,
                              hipStream_t stream) {
    (void)n_in; (void)d_ws; (void)ws_size;

    const float*         coords  = (const float*)d_in[0];        // (B,N,3)
    const unsigned char* active  = (const unsigned char*)d_in[1];// (B,N) bool8
    const float*         occ     = (const float*)d_in[2];        // (B,N)
    const float*         lmax    = (const float*)d_in[3];        // (3,)
    const float*         radial  = (const float*)d_in[4];        // (B,N,R)
    const float*         gmat    = (const float*)d_in[5];        // (3,3)
    float*               out     = (float*)d_out;

    const int atoms_total = in_sizes[2];                 // B*N
    const int B           = out_size / GRID_VOX;         // out = B*128^3
    const int N           = atoms_total / (B > 0 ? B : 1);
    const int n_radial    = atoms_total > 0 ? in_sizes[4] / atoms_total : 0;

    // output is accumulated via atomics -> must start from zero every call
    (void)hipMemsetAsync(d_out, 0, (size_t)out_size * sizeof(float), stream);

    splat_density_kernel<<<dim3(atoms_total), dim3(BLOCK), 0, stream>>>(
        coords, active, occ, lmax, radial, gmat, out, N, n_radial);
}